// NodeModel_47270410059817
// MI455X (gfx1250) — compile-verified
//
#include <hip/hip_runtime.h>
#include <math.h>

typedef __attribute__((ext_vector_type(2))) float v2f;
typedef __attribute__((ext_vector_type(8))) float v8f;

#define NODE_F    256
#define EDGE_F    96
#define GLOB_F    64
#define HIDDEN    1024
#define MLP_IN    608

#define TM 128
#define TN 64
#define TK 32
#define AW (TK + 4)        // A tile row stride (words): 36 -> 144B, 16B aligned
#define BW (2 * TN + 32)   // B pair-row stride (words): 160 ≡ 32 (mod 64 banks)

// -------- float atomic max via monotonic integer atomics --------
__device__ __forceinline__ void atomicMaxF(float* addr, float val) {
    if (val >= 0.0f)
        atomicMax((int*)addr, __float_as_int(val));
    else
        atomicMin((unsigned int*)addr, __float_as_uint(val));
}

// -------- init workspace accumulators --------
__global__ void init_kernel(float* __restrict__ out1, float* __restrict__ out2,
                            float* __restrict__ cnt, int n96, int n) {
    int i = blockIdx.x * blockDim.x + threadIdx.x;
    if (i < n96) {
        out1[i] = 0.0f;
        out2[i] = -__builtin_inff();
    }
    if (i < n) cnt[i] = 0.0f;
}

// -------- edge scatter: sum / max / count per destination node --------
__global__ void scatter_kernel(const float* __restrict__ ea, const int* __restrict__ dstIdx,
                               float* __restrict__ out1, float* __restrict__ out2,
                               float* __restrict__ cnt, int total) {
    int idx = blockIdx.x * blockDim.x + threadIdx.x;
    if (idx >= total) return;
    int e = idx / EDGE_F;
    int f = idx - e * EDGE_F;
    int d = dstIdx[e];
    float v = ea[idx];
    atomicAdd(&out1[d * EDGE_F + f], v);
    atomicMaxF(&out2[d * EDGE_F + f], v);
    if (f == 0) atomicAdd(&cnt[d], 1.0f);
}

// -------- build H = [x | sum | max(where cnt>0) | mean | u[batch]]  [N, 608] --------
__global__ void assemble_kernel(const float* __restrict__ x, const float* __restrict__ u,
                                const int* __restrict__ batch,
                                const float* __restrict__ out1, const float* __restrict__ out2,
                                const float* __restrict__ cnt, float* __restrict__ H) {
    int i = blockIdx.x;
    float c = cnt[i];
    float inv = 1.0f / fmaxf(c, 1.0f);
    int g = batch[i];
    for (int f = threadIdx.x; f < MLP_IN; f += blockDim.x) {
        float v;
        if (f < 256) {
            v = x[(size_t)i * 256 + f];
        } else if (f < 352) {
            v = out1[(size_t)i * EDGE_F + (f - 256)];
        } else if (f < 448) {
            float m = out2[(size_t)i * EDGE_F + (f - 352)];
            v = (c > 0.0f) ? m : 0.0f;
        } else if (f < 544) {
            v = out1[(size_t)i * EDGE_F + (f - 448)] * inv;
        } else {
            v = u[(size_t)g * GLOB_F + (f - 544)];
        }
        H[(size_t)i * MLP_IN + f] = v;
    }
}

// -------- fp32 WMMA GEMM, register-staged double buffering --------
// Block tile 128x64, 8 waves (wave32), wave = one 16-row strip x four 16x16 WMMA tiles.
// B staged K-pair interleaved: Bs[k/2][2n + (k&1)] so a fragment is one ds_load_b64.
template <bool RELU, bool RESID>
__global__ __launch_bounds__(256) void gemm_kernel(const float* __restrict__ A, int K,
                                                   const float* __restrict__ B, int ldb,
                                                   const float* __restrict__ bias,
                                                   const float* __restrict__ resid,
                                                   float* __restrict__ C, int ldc, int nRows) {
    __shared__ float As[2][TM][AW];
    __shared__ float Bs[2][TK / 2][BW];

    const int tid    = threadIdx.x;
    const int wave   = tid >> 5;
    const int lane   = tid & 31;
    const int laneLo = lane & 15;
    const int laneHi = lane >> 4;  // 0 or 1

    const int rowBlk = blockIdx.y * TM;
    const int colBlk = blockIdx.x * TN;

    v8f acc[4];
#pragma unroll
    for (int t = 0; t < 4; ++t)
#pragma unroll
        for (int r = 0; r < 8; ++r) acc[t][r] = 0.0f;

    // global->register staging assignments (256 threads)
    const int arow  = tid >> 1;         // 0..127, 2 threads per row
    const int acol0 = (tid & 1) * 16;   // 16 floats (4x float4)
    const int brow  = tid >> 3;         // 0..31, 8 threads per row
    const int bcol0 = (tid & 7) * 8;    // 8 floats (2x float4)

    const int wr = wave * 16;
    const int grow = rowBlk + arow;
    const bool aValid = grow < nRows;
    const float* aSrcBase = A + (size_t)(aValid ? grow : 0) * K + acol0;
    const float* bSrcBase = B + (size_t)brow * ldb + colBlk + bcol0;

    float4 av[4], bv[2];

    // ---- prologue: fetch chunk 0 into registers, stage into buffer 0 ----
    {
        if (aValid) {
            av[0] = ((const float4*)aSrcBase)[0];
            av[1] = ((const float4*)aSrcBase)[1];
            av[2] = ((const float4*)aSrcBase)[2];
            av[3] = ((const float4*)aSrcBase)[3];
        } else {
            float4 z = make_float4(0.f, 0.f, 0.f, 0.f);
            av[0] = av[1] = av[2] = av[3] = z;
        }
        bv[0] = ((const float4*)bSrcBase)[0];
        bv[1] = ((const float4*)bSrcBase)[1];

        ((float4*)&As[0][arow][acol0])[0] = av[0];
        ((float4*)&As[0][arow][acol0])[1] = av[1];
        ((float4*)&As[0][arow][acol0])[2] = av[2];
        ((float4*)&As[0][arow][acol0])[3] = av[3];
        float* d = &Bs[0][brow >> 1][brow & 1];
        d[(bcol0 + 0) * 2] = bv[0].x;
        d[(bcol0 + 1) * 2] = bv[0].y;
        d[(bcol0 + 2) * 2] = bv[0].z;
        d[(bcol0 + 3) * 2] = bv[0].w;
        d[(bcol0 + 4) * 2] = bv[1].x;
        d[(bcol0 + 5) * 2] = bv[1].y;
        d[(bcol0 + 6) * 2] = bv[1].z;
        d[(bcol0 + 7) * 2] = bv[1].w;
    }
    __syncthreads();

    const int nCh = K / TK;  // K is a multiple of 32 (608, 1024)
    for (int kc = 0; kc < nCh; ++kc) {
        const int cur  = kc & 1;
        const int nxt  = cur ^ 1;
        const bool more = (kc + 1) < nCh;

        // ---- issue global loads for next chunk (latency hidden under WMMAs) ----
        if (more) {
            const int k0n = (kc + 1) * TK;
            if (aValid) {
                const float* src = aSrcBase + k0n;
                av[0] = ((const float4*)src)[0];
                av[1] = ((const float4*)src)[1];
                av[2] = ((const float4*)src)[2];
                av[3] = ((const float4*)src)[3];
            }
            const float* bsrc = bSrcBase + (size_t)k0n * ldb;
            bv[0] = ((const float4*)bsrc)[0];
            bv[1] = ((const float4*)bsrc)[1];
        }

        // ---- 8 k-steps of V_WMMA_F32_16X16X4_F32 from buffer `cur` ----
#pragma unroll
        for (int kk = 0; kk < TK; kk += 4) {
            v2f a = *(const v2f*)&As[cur][wr + laneLo][kk + laneHi * 2];
#pragma unroll
            for (int t = 0; t < 4; ++t) {
                v2f b = *(const v2f*)&Bs[cur][(kk >> 1) + laneHi][(t * 16 + laneLo) * 2];
                acc[t] = __builtin_amdgcn_wmma_f32_16x16x4_f32(
                    /*neg_a=*/false, a, /*neg_b=*/false, b,
                    /*c_mod=*/(short)0, acc[t], /*reuse_a=*/false, /*reuse_b=*/false);
            }
        }

        // ---- stage next chunk into the other buffer ----
        if (more) {
            ((float4*)&As[nxt][arow][acol0])[0] = av[0];
            ((float4*)&As[nxt][arow][acol0])[1] = av[1];
            ((float4*)&As[nxt][arow][acol0])[2] = av[2];
            ((float4*)&As[nxt][arow][acol0])[3] = av[3];
            float* d = &Bs[nxt][brow >> 1][brow & 1];
            d[(bcol0 + 0) * 2] = bv[0].x;
            d[(bcol0 + 1) * 2] = bv[0].y;
            d[(bcol0 + 2) * 2] = bv[0].z;
            d[(bcol0 + 3) * 2] = bv[0].w;
            d[(bcol0 + 4) * 2] = bv[1].x;
            d[(bcol0 + 5) * 2] = bv[1].y;
            d[(bcol0 + 6) * 2] = bv[1].z;
            d[(bcol0 + 7) * 2] = bv[1].w;
        }
        __syncthreads();
    }

    // ---- epilogue: C vgpr r, lane l -> row (l/16)*8 + r, col l%16 ----
#pragma unroll
    for (int t = 0; t < 4; ++t) {
        int col   = colBlk + t * 16 + laneLo;
        float bvv = bias[col];
        int rbase = rowBlk + wr + laneHi * 8;
#pragma unroll
        for (int r = 0; r < 8; ++r) {
            int row = rbase + r;
            if (row < nRows) {
                float v = acc[t][r] + bvv;
                if (RELU) v = fmaxf(v, 0.0f);
                if (RESID) v += resid[(size_t)row * ldc + col];
                C[(size_t)row * ldc + col] = v;
            }
        }
    }
}

extern "C" void kernel_launch(void* const* d_in, const int* in_sizes, int n_in,
                              void* d_out, int out_size, void* d_ws, size_t ws_size,
                              hipStream_t stream) {
    const float* x    = (const float*)d_in[0];
    const float* ea   = (const float*)d_in[1];
    const float* u    = (const float*)d_in[2];
    const float* W1   = (const float*)d_in[3];
    const float* b1   = (const float*)d_in[4];
    const float* W2   = (const float*)d_in[5];
    const float* b2   = (const float*)d_in[6];
    const int*   eidx = (const int*)d_in[7];
    const int*   batch= (const int*)d_in[8];
    float* out = (float*)d_out;

    const int N = in_sizes[0] / NODE_F;   // 50000
    const int E = in_sizes[1] / EDGE_F;   // 800000
    const int* dstIdx = eidx + E;         // edge_index[1]

    // workspace layout (floats)
    float* ws   = (float*)d_ws;
    float* out1 = ws;                               // N*96
    float* out2 = out1 + (size_t)N * EDGE_F;        // N*96
    float* cnt  = out2 + (size_t)N * EDGE_F;        // N
    float* H    = cnt + N;                          // N*608
    float* h1   = H + (size_t)N * MLP_IN;           // N*1024

    // 1) init accumulators
    {
        int n96 = N * EDGE_F;
        int blocks = (n96 + 255) / 256;
        init_kernel<<<blocks, 256, 0, stream>>>(out1, out2, cnt, n96, N);
    }
    // 2) edge scatter
    {
        int total = E * EDGE_F;  // 76.8M < 2^31
        int blocks = (total + 255) / 256;
        scatter_kernel<<<blocks, 256, 0, stream>>>(ea, dstIdx, out1, out2, cnt, total);
    }
    // 3) assemble H
    assemble_kernel<<<N, 256, 0, stream>>>(x, u, batch, out1, out2, cnt, H);

    // 4) GEMM1: h1 = relu(H @ W1 + b1)   [N,608]x[608,1024]
    {
        dim3 grid(HIDDEN / TN, (N + TM - 1) / TM);
        gemm_kernel<true, false><<<grid, 256, 0, stream>>>(H, MLP_IN, W1, HIDDEN, b1,
                                                           nullptr, h1, HIDDEN, N);
    }
    // 5) GEMM2: out = h1 @ W2 + b2 + x   [N,1024]x[1024,256]
    {
        dim3 grid(NODE_F / TN, (N + TM - 1) / TM);
        gemm_kernel<false, true><<<grid, 256, 0, stream>>>(h1, HIDDEN, W2, NODE_F, b2,
                                                           x, out, NODE_F, N);
    }
}